// Point_Embedding_66331474919943
// MI455X (gfx1250) — compile-verified
//
#include <hip/hip_runtime.h>
#include <math.h>

// ---- problem constants (from reference) ----
#define B_   16
#define C_   3
#define N_   2048
#define K_   32
#define CH   64
#define NPTS (B_ * N_)          // 32768 points
#define MTOT (B_ * N_ * K_)     // 1048576 BN samples
#define NEG_SLOPE 0.2f
#define BN_EPS    1e-5f

typedef __attribute__((ext_vector_type(2))) float v2f;
typedef __attribute__((ext_vector_type(8))) float v8f;

// CDNA5 fp32 WMMA: D(16x16,f32) = A(16x4,f32) x B(4x16,f32) + C
static __device__ __forceinline__ v8f wmma_f32(v2f a, v2f b, v8f c) {
  return __builtin_amdgcn_wmma_f32_16x16x4_f32(
      /*neg_a=*/false, a, /*neg_b=*/false, b,
      /*c_mod=*/(short)0, c, /*reuse_a=*/false, /*reuse_b=*/false);
}

// ---- workspace float-region layout (after NPTS*K_ ints of idx) ----
#define OF_SUMX   0      // 6
#define OF_SUMXX  6      // 36
#define OF_SC1    42     // 64  (gamma*inv_std)
#define OF_SH1    106    // 64  (beta - mu*scale)
#define OF_SUMH   170    // 64
#define OF_SUMHH  234    // 4096
#define OF_SC2    4330   // 64
#define OF_SH2    4394   // 64
#define F_TOTAL   4458

// =====================================================================
// Kernel 1: brute-force kNN (32 nearest b-points for each a-point)
// One block per (n, b); distances in LDS; 32 rounds of block argmin.
// =====================================================================
__global__ __launch_bounds__(256) void knn_kernel(const float* __restrict__ a,
                                                  const float* __restrict__ b,
                                                  int* __restrict__ idx) {
  __shared__ float dist[N_];
  __shared__ float rmin[256];
  __shared__ int   rind[256];
  const int tid = threadIdx.x;
  const int n = blockIdx.x, bi = blockIdx.y;
  const float* ab = a + (size_t)bi * C_ * N_;
  const float* bb = b + (size_t)bi * C_ * N_;
  const float ax = ab[n], ay = ab[N_ + n], az = ab[2 * N_ + n];
  for (int m = tid; m < N_; m += 256) {
    float dx = ax - bb[m], dy = ay - bb[N_ + m], dz = az - bb[2 * N_ + m];
    dist[m] = dx * dx + dy * dy + dz * dz;
  }
  __syncthreads();
  for (int jj = 0; jj < K_; jj++) {
    float best = 3.4e38f; int bm = N_;
    for (int m = tid; m < N_; m += 256) {
      float d = dist[m];
      if (d < best) { best = d; bm = m; }   // strided -> earliest m wins ties
    }
    rmin[tid] = best; rind[tid] = bm;
    __syncthreads();
    for (int s = 128; s > 0; s >>= 1) {
      if (tid < s) {
        float o = rmin[tid + s];
        if (o < rmin[tid] || (o == rmin[tid] && rind[tid + s] < rind[tid])) {
          rmin[tid] = o; rind[tid] = rind[tid + s];
        }
      }
      __syncthreads();
    }
    if (tid == 0) {
      idx[((size_t)(bi * N_ + n)) * K_ + jj] = rind[0];
      dist[rind[0]] = 3.4e38f;
    }
    __syncthreads();
  }
}

// =====================================================================
// Kernel 2: moments of x = [a_j - a_i, a_i] (6-vec): sum x, sum x x^T
// =====================================================================
__global__ __launch_bounds__(256) void stats_x_kernel(const float* __restrict__ a,
                                                      const int* __restrict__ idx,
                                                      float* __restrict__ F) {
  const int p = blockIdx.x * blockDim.x + threadIdx.x;  // exactly NPTS threads
  float sx[6] = {0, 0, 0, 0, 0, 0};
  float sxx[36];
#pragma unroll
  for (int i = 0; i < 36; i++) sxx[i] = 0.0f;
  if (p < NPTS) {
    const int bi = p / N_, n = p % N_;
    const float* ab = a + (size_t)bi * C_ * N_;
    float x[6];
    x[3] = ab[n]; x[4] = ab[N_ + n]; x[5] = ab[2 * N_ + n];
    for (int k = 0; k < K_; k++) {
      const int j = idx[(size_t)p * K_ + k];
      x[0] = ab[j] - x[3]; x[1] = ab[N_ + j] - x[4]; x[2] = ab[2 * N_ + j] - x[5];
#pragma unroll
      for (int c = 0; c < 6; c++) {
        sx[c] += x[c];
#pragma unroll
        for (int d = 0; d < 6; d++) sxx[c * 6 + d] += x[c] * x[d];
      }
    }
  }
  // wave32 reduction, lane0 atomics
#pragma unroll
  for (int i = 0; i < 6; i++) {
    float r = sx[i];
    for (int s = 16; s > 0; s >>= 1) r += __shfl_down(r, s, 32);
    if ((threadIdx.x & 31) == 0) atomicAdd(&F[OF_SUMX + i], r);
  }
#pragma unroll
  for (int i = 0; i < 36; i++) {
    float r = sxx[i];
    for (int s = 16; s > 0; s >>= 1) r += __shfl_down(r, s, 32);
    if ((threadIdx.x & 31) == 0) atomicAdd(&F[OF_SUMXX + i], r);
  }
}

// =====================================================================
// Kernel 3: derive BN1 scale/shift analytically: y1 = x W1^T
// mu = w.E[x];  E[y^2] = w^T E[xx^T] w
// =====================================================================
__global__ void bn1_stats_kernel(const float* __restrict__ W1,
                                 const float* __restrict__ g1,
                                 const float* __restrict__ b1,
                                 float* __restrict__ F) {
  const int o = threadIdx.x;
  if (o >= CH) return;
  const float invM = 1.0f / (float)MTOT;
  float mx[6], wv[6];
#pragma unroll
  for (int c = 0; c < 6; c++) { mx[c] = F[OF_SUMX + c] * invM; wv[c] = W1[o * 6 + c]; }
  float mu = 0.0f;
#pragma unroll
  for (int c = 0; c < 6; c++) mu += wv[c] * mx[c];
  float ey2 = 0.0f;
#pragma unroll
  for (int c = 0; c < 6; c++)
#pragma unroll
    for (int d = 0; d < 6; d++) ey2 += wv[c] * wv[d] * (F[OF_SUMXX + c * 6 + d] * invM);
  const float var = ey2 - mu * mu;
  const float sc = g1[o] * rsqrtf(var + BN_EPS);
  F[OF_SC1 + o] = sc;
  F[OF_SH1 + o] = b1[o] - mu * sc;
}

// ---- shared helpers -------------------------------------------------
static __device__ __forceinline__ void stage_x(const float* __restrict__ a,
                                               const int* __restrict__ idx,
                                               int bi, int n, float* Xs, int tid) {
  if (tid < K_) {
    const float* ab = a + (size_t)bi * C_ * N_;
    const float ax = ab[n], ay = ab[N_ + n], az = ab[2 * N_ + n];
    const int j = idx[((size_t)(bi * N_ + n)) * K_ + tid];
    float* r = Xs + tid * 8;
    r[0] = ab[j] - ax; r[1] = ab[N_ + j] - ay; r[2] = ab[2 * N_ + j] - az;
    r[3] = ax; r[4] = ay; r[5] = az; r[6] = 0.0f; r[7] = 0.0f;   // pad K 6->8
  }
}

// GEMM1 (32x8 @ 8x64) via fp32 WMMA, fused BN1 + LeakyReLU, result -> Hs[32][64]
static __device__ __forceinline__ void gemm1_bn(const float* Xs, const float* W1s,
                                                const float* s1, const float* sh1,
                                                float* Hs, int tid) {
  const int w = tid >> 5, lane = tid & 31, l = lane & 15;
  const bool hi = lane >= 16;
  const int m0 = (w >> 2) * 16, n0 = (w & 3) * 16;
  v8f acc = {};
#pragma unroll
  for (int c0 = 0; c0 < 8; c0 += 4) {
    const int kk = c0 + (hi ? 2 : 0);
    v2f Av = { Xs[(m0 + l) * 8 + kk], Xs[(m0 + l) * 8 + kk + 1] };   // A[M=row][K]
    v2f Bv = { W1s[(n0 + l) * 8 + kk], W1s[(n0 + l) * 8 + kk + 1] }; // B[K][N]=W1^T
    acc = wmma_f32(Av, Bv, acc);
  }
  const int col = n0 + l;
  const float sc = s1[col], sh = sh1[col];
  const int rbase = m0 + (hi ? 8 : 0);
#pragma unroll
  for (int v = 0; v < 8; v++) {
    float y = acc[v] * sc + sh;
    y = y > 0.0f ? y : NEG_SLOPE * y;
    Hs[(rbase + v) * CH + col] = y;
  }
}

// =====================================================================
// Kernel 4: moments of h = lrelu(bn1(x W1^T)): sum h (64), sum h h^T
// (64x64 SYRK per point done with WMMA, accumulated in registers)
// =====================================================================
__global__ __launch_bounds__(256) void stats_h_kernel(const float* __restrict__ a,
                                                      const int* __restrict__ idx,
                                                      const float* __restrict__ W1,
                                                      float* __restrict__ F) {
  __shared__ float Xs[K_ * 8];
  __shared__ float W1s[CH * 8];
  __shared__ float Hs[K_ * CH];
  __shared__ float s1[CH], sh1[CH];
  const int tid = threadIdx.x;
  for (int i = tid; i < CH * 8; i += 256) {
    const int o = i >> 3, c = i & 7;
    W1s[i] = (c < 6) ? W1[o * 6 + c] : 0.0f;
  }
  if (tid < CH) { s1[tid] = F[OF_SC1 + tid]; sh1[tid] = F[OF_SH1 + tid]; }
  __syncthreads();

  const int w = tid >> 5, lane = tid & 31, l = lane & 15;
  const bool hi = lane >= 16;
  v8f accT[2] = {};
  float acc_h = 0.0f;

  for (int p = blockIdx.x; p < NPTS; p += gridDim.x) {   // uniform trip count
    const int bi = p / N_, n = p % N_;
    stage_x(a, idx, bi, n, Xs, tid);
    __syncthreads();
    gemm1_bn(Xs, W1s, s1, sh1, Hs, tid);
    __syncthreads();
    if (tid < CH) {
      float s = 0.0f;
#pragma unroll
      for (int k = 0; k < K_; k++) s += Hs[k * CH + tid];
      acc_h += s;
    }
    // H^T H  (M=64 x N=64, K=32): 16 tiles, 2 per wave, WMMA fp32
#pragma unroll
    for (int t = 0; t < 2; t++) {
      const int tt = 2 * w + t;
      const int m0 = (tt >> 2) * 16, n0 = (tt & 3) * 16;
      v8f acc = accT[t];
#pragma unroll
      for (int k0 = 0; k0 < K_; k0 += 4) {
        const int kk = k0 + (hi ? 2 : 0);
        v2f Av = { Hs[kk * CH + m0 + l], Hs[(kk + 1) * CH + m0 + l] }; // A=H^T
        v2f Bv = { Hs[kk * CH + n0 + l], Hs[(kk + 1) * CH + n0 + l] }; // B=H
        acc = wmma_f32(Av, Bv, acc);
      }
      accT[t] = acc;
    }
    __syncthreads();
  }
  if (tid < CH) atomicAdd(&F[OF_SUMH + tid], acc_h);
#pragma unroll
  for (int t = 0; t < 2; t++) {
    const int tt = 2 * w + t;
    const int m0 = (tt >> 2) * 16, n0 = (tt & 3) * 16;
    const int rbase = m0 + (hi ? 8 : 0);
#pragma unroll
    for (int v = 0; v < 8; v++)
      atomicAdd(&F[OF_SUMHH + (rbase + v) * CH + (n0 + l)], accT[t][v]);
  }
}

// =====================================================================
// Kernel 5: derive BN2 scale/shift: y2 = h W2^T
// =====================================================================
__global__ void bn2_stats_kernel(const float* __restrict__ W2,
                                 const float* __restrict__ g2,
                                 const float* __restrict__ b2,
                                 float* __restrict__ F) {
  __shared__ float mh[CH];
  const int o = threadIdx.x;
  const float invM = 1.0f / (float)MTOT;
  if (o < CH) mh[o] = F[OF_SUMH + o] * invM;
  __syncthreads();
  if (o >= CH) return;
  float mu = 0.0f;
  for (int c = 0; c < CH; c++) mu += W2[o * CH + c] * mh[c];
  float ey2 = 0.0f;
  for (int c = 0; c < CH; c++) {
    const float wc = W2[o * CH + c];
    for (int d = 0; d < CH; d++)
      ey2 += wc * W2[o * CH + d] * (F[OF_SUMHH + c * CH + d] * invM);
  }
  const float var = ey2 - mu * mu;
  const float sc = g2[o] * rsqrtf(var + BN_EPS);
  F[OF_SC2 + o] = sc;
  F[OF_SH2 + o] = b2[o] - mu * sc;
}

// =====================================================================
// Kernel 6: fully fused per point: gather -> GEMM1+BN1+lrelu ->
// GEMM2+BN2+lrelu -> max over K -> out[b][o][n]
// =====================================================================
__global__ __launch_bounds__(256) void final_kernel(const float* __restrict__ a,
                                                    const int* __restrict__ idx,
                                                    const float* __restrict__ W1,
                                                    const float* __restrict__ W2,
                                                    const float* __restrict__ F,
                                                    float* __restrict__ out) {
  __shared__ float Xs[K_ * 8];
  __shared__ float W1s[CH * 8];
  __shared__ float W2s[CH * CH];
  __shared__ float Hs[K_ * CH];
  __shared__ float s1[CH], sh1[CH], s2[CH], sh2[CH];
  __shared__ float partial[2][CH];
  const int tid = threadIdx.x;
  const int n = blockIdx.x, bi = blockIdx.y;

  for (int i = tid; i < CH * 8; i += 256) {
    const int o = i >> 3, c = i & 7;
    W1s[i] = (c < 6) ? W1[o * 6 + c] : 0.0f;
  }
  for (int i = tid; i < CH * CH; i += 256) W2s[i] = W2[i];
  if (tid < CH) {
    s1[tid] = F[OF_SC1 + tid]; sh1[tid] = F[OF_SH1 + tid];
    s2[tid] = F[OF_SC2 + tid]; sh2[tid] = F[OF_SH2 + tid];
  }
  stage_x(a, idx, bi, n, Xs, tid);
  __syncthreads();
  gemm1_bn(Xs, W1s, s1, sh1, Hs, tid);
  __syncthreads();

  // GEMM2: (32x64) @ (64x64)^T, 8 tiles, 1 per wave, K-loop 16 WMMAs
  const int w = tid >> 5, lane = tid & 31, l = lane & 15;
  const bool hi = lane >= 16;
  const int m0 = (w >> 2) * 16, n0 = (w & 3) * 16;
  v8f acc = {};
#pragma unroll
  for (int c0 = 0; c0 < CH; c0 += 4) {
    const int kk = c0 + (hi ? 2 : 0);
    v2f Av = { Hs[(m0 + l) * CH + kk], Hs[(m0 + l) * CH + kk + 1] };
    v2f Bv = { W2s[(n0 + l) * CH + kk], W2s[(n0 + l) * CH + kk + 1] };
    acc = wmma_f32(Av, Bv, acc);
  }
  const int col = n0 + l;
  const float sc = s2[col], sh = sh2[col];
  float m = -3.4e38f;
#pragma unroll
  for (int v = 0; v < 8; v++) {
    float y = acc[v] * sc + sh;
    y = y > 0.0f ? y : NEG_SLOPE * y;
    m = fmaxf(m, y);
  }
  m = fmaxf(m, __shfl_xor(m, 16, 32));    // merge rows m0..m0+7 with m0+8..m0+15
  if (lane < 16) partial[w >> 2][col] = m;
  __syncthreads();
  if (tid < CH)
    out[((size_t)bi * CH + tid) * N_ + n] = fmaxf(partial[0][tid], partial[1][tid]);
}

// =====================================================================
extern "C" void kernel_launch(void* const* d_in, const int* in_sizes, int n_in,
                              void* d_out, int out_size, void* d_ws, size_t ws_size,
                              hipStream_t stream) {
  const float* a  = (const float*)d_in[0];
  const float* b  = (const float*)d_in[1];
  const float* W1 = (const float*)d_in[2];
  const float* g1 = (const float*)d_in[3];
  const float* b1 = (const float*)d_in[4];
  const float* W2 = (const float*)d_in[5];
  const float* g2 = (const float*)d_in[6];
  const float* b2 = (const float*)d_in[7];
  float* out = (float*)d_out;

  int*   idx = (int*)d_ws;                                     // 4 MB
  float* F   = (float*)((char*)d_ws + (size_t)NPTS * K_ * sizeof(int));

  hipMemsetAsync(F, 0, F_TOTAL * sizeof(float), stream);       // zero accumulators

  knn_kernel    <<<dim3(N_, B_), 256, 0, stream>>>(a, b, idx);
  stats_x_kernel<<<NPTS / 256, 256, 0, stream>>>(a, idx, F);
  bn1_stats_kernel<<<1, 64, 0, stream>>>(W1, g1, b1, F);
  stats_h_kernel<<<512, 256, 0, stream>>>(a, idx, W1, F);
  bn2_stats_kernel<<<1, 64, 0, stream>>>(W2, g2, b2, F);
  final_kernel  <<<dim3(N_, B_), 256, 0, stream>>>(a, idx, W1, W2, F, out);
}